// TWC_2894807958278
// MI455X (gfx1250) — compile-verified
//
#include <hip/hip_runtime.h>
#include <cmath>

// ---------------------------------------------------------------------------
// TWC recurrent network, MI455X (gfx1250, wave32).
// Memory-bound (~310 MB traffic => ~13us @ 23.3 TB/s HBM); GEMM FLOPs are
// tiny (2.1 GFLOP), so we use the EXACT fp32 WMMA (v_wmma_f32_16x16x4_f32).
// The 4-step linear recurrence collapses to E4 = a^4*E0 + (1+a+a^2+a^3)*I
// with a = dec_hid - deg.
//
// Weight tables are stored K-PAIR-INTERLEAVED so each WMMA B-fragment is one
// contiguous 8 bytes in LDS -> single ds_load_b64 into an even-aligned VGPR
// pair (no v_mov repacking in the inner loop):
//   W[kk][n]  ->  ws[base + ((kk>>1)*N + n)*2 + (kk&1)]
// ---------------------------------------------------------------------------

typedef __attribute__((ext_vector_type(2))) float v2f;
typedef __attribute__((ext_vector_type(8))) float v8f;

#define WS_WCOMB 0      // [64][64] interleaved: sp(w_hid_EX) - sp(w_hid_IN)
#define WS_CIN   4096   // [4][64]  interleaved: M - sp(w_in2hid_IN)
#define WS_W2    4352   // [64][16] interleaved: sp(w_hid2out) zero-padded
#define WS_A4    5376   // [64]     a^4
#define WS_S     5440   // [64]     1 + a + a^2 + a^3
#define WS_TH    5504   // [64]     th_hid copy
#define WS_TOTAL 5568   // floats (22272 bytes)

__device__ __forceinline__ float splus(float v) {
    return (v > 20.0f) ? v : log1pf(expf(v));
}

// ---------------------------------------------------------------------------
// Prep: batch-invariant tables. One block, deterministic (no float atomics).
// ---------------------------------------------------------------------------
__global__ void twc_prep(const float* __restrict__ w_in2hid,
                         const float* __restrict__ w_hid_IN,
                         const float* __restrict__ w_hid_EX,
                         const float* __restrict__ w_hid2out,
                         const float* __restrict__ gj_w,
                         const float* __restrict__ dec_hid,
                         const float* __restrict__ th_hid,
                         const int*   __restrict__ gj_src,
                         const int*   __restrict__ gj_dst,
                         float* __restrict__ ws) {
    int tid = threadIdx.x;
    for (int i = tid; i < 4096; i += 256) {
        int k = i >> 6, n = i & 63;
        ws[WS_WCOMB + (((k >> 1) * 64 + n) << 1) + (k & 1)] =
            splus(w_hid_EX[i]) - splus(w_hid_IN[i]);
    }
    {   // Cin base (256 = 4*64 elements, one per thread)
        int k = tid >> 6, n = tid & 63;
        ws[WS_CIN + (((k >> 1) * 64 + n) << 1) + (k & 1)] = -splus(w_in2hid[tid]);
    }
    for (int i = tid; i < 1024; i += 256) {
        int k = i >> 4, n = i & 15;
        ws[WS_W2 + (((k >> 1) * 16 + n) << 1) + (k & 1)] =
            (n < 8) ? splus(w_hid2out[k * 8 + n]) : 0.0f;
    }
    if (tid < 64) ws[WS_TH + tid] = th_hid[tid];
    __syncthreads();
    // Gap-junction scatter: one thread per destination column -> deterministic
    if (tid < 64) {
        float deg = 0.0f;
        for (int e = 0; e < 128; ++e) {
            if (gj_dst[e] == tid) {
                float sw = splus(gj_w[e]);
                deg += sw;
                int s = gj_src[e];   // column-private RMW (only thread `tid`)
                ws[WS_CIN + (((s >> 1) * 64 + tid) << 1) + (s & 1)] += sw;
            }
        }
        float a  = dec_hid[tid] - deg;
        float a2 = a * a;
        ws[WS_A4 + tid] = a2 * a2;                 // a^4
        ws[WS_S  + tid] = 1.0f + a + a2 + a2 * a;  // 1+a+a^2+a^3
    }
}

// ---------------------------------------------------------------------------
// Main: 8 waves/block, 16 batch rows per wave.
// Per wave: 84 x v_wmma_f32_16x16x4_f32 (exact fp32).
// ---------------------------------------------------------------------------
__global__ __launch_bounds__(256) void twc_main(
    const float* __restrict__ x,
    const float* __restrict__ hid_E0,
    const float* __restrict__ hid_O0,
    const float* __restrict__ out_E0,
    const float* __restrict__ th_in,  const float* __restrict__ dec_in,
    const float* __restrict__ th_out, const float* __restrict__ dec_out,
    const float* __restrict__ ws,
    float* __restrict__ out, int B)
{
    __shared__ v2f smv[WS_TOTAL / 2];
    float* sm = (float*)smv;
    for (int i = threadIdx.x; i < WS_TOTAL; i += 256) sm[i] = ws[i];
    __syncthreads();

    const v2f* WI   = (const v2f*)(sm + WS_WCOMB);  // [32 pairs][64]
    const v2f* CinI = (const v2f*)(sm + WS_CIN);    // [2 pairs][64]
    const v2f* W2I  = (const v2f*)(sm + WS_W2);     // [32 pairs][16]
    const float* a4L = sm + WS_A4;
    const float* sL  = sm + WS_S;
    const float* thL = sm + WS_TH;

    const int  wave = threadIdx.x >> 5;
    const int  lane = threadIdx.x & 31;
    const long rowbase = (long)(blockIdx.x * 8 + wave) * 16;
    if (rowbase + 16 > B) return;   // wave-uniform: EXEC stays all-1s for WMMA

    // Output sections (flat concatenation in reference return order)
    float* o_act = out;
    float* o_Ein = out + (long)B * 8;
    float* o_Oin = out + (long)B * 12;
    float* o_Eh  = out + (long)B * 16;
    float* o_Oh  = out + (long)B * 80;
    float* o_Eo  = out + (long)B * 144;
    float* o_Oo  = out + (long)B * 152;

    const int m    = lane & 15;      // A: row / B: col / C,D: col
    const int half = lane >> 4;
    const int koff = half * 2;       // K pair selected by lane half
    const int n    = m;

    // ---- E_in / O_in: 16x4 elementwise, fully coalesced ----
    {
        long base = rowbase * 4;
        #pragma unroll
        for (int rep = 0; rep < 2; ++rep) {
            int   e  = lane + rep * 32;
            int   c  = e & 3;
            float xv = x[base + e];
            float Ei = dec_in[c] * xv;
            o_Ein[base + e] = Ei;
            o_Oin[base + e] = (Ei >= th_in[c]) ? Ei : 0.0f;
        }
    }

    // ---- Seed accumulators with x @ Cin (single K=4 WMMA per tile) ----
    v8f accI[4];
    {
        float2 t = *(const float2*)(x + (rowbase + m) * 4 + koff);
        v2f ax; ax[0] = t.x; ax[1] = t.y;
        #pragma unroll
        for (int j = 0; j < 4; ++j) {
            v2f bx = CinI[half * 64 + j * 16 + n];    // single ds_load_b64
            v8f cz = {};
            accI[j] = __builtin_amdgcn_wmma_f32_16x16x4_f32(
                false, ax, false, bx, (short)0, cz, false, false);
        }
    }
    v8f accO = {};

    // ---- Main K loop: hid_O0[16,64] @ W_comb[64,64] and @ W2pad[64,16] ----
    const float* Arow = hid_O0 + (rowbase + m) * 64;
    #pragma unroll
    for (int k = 0; k < 16; ++k) {
        const int p = k * 2 + half;                   // K-pair index
        float2 t = *(const float2*)(Arow + k * 4 + koff);  // global_load_b64
        v2f a; a[0] = t.x; a[1] = t.y;
        #pragma unroll
        for (int j = 0; j < 4; ++j) {
            v2f b = WI[p * 64 + j * 16 + n];          // single ds_load_b64
            accI[j] = __builtin_amdgcn_wmma_f32_16x16x4_f32(
                false, a, false, b, (short)0, accI[j], false, false);
        }
        v2f bo = W2I[p * 16 + n];
        accO = __builtin_amdgcn_wmma_f32_16x16x4_f32(
            false, a, false, bo, (short)0, accO, false, false);
    }

    // ---- Hidden epilogue: closed-form recurrence + gate ----
    #pragma unroll
    for (int j = 0; j < 4; ++j) {
        const int   ng  = j * 16 + n;
        const float a4c = a4L[ng], sc = sL[ng], thc = thL[ng];
        #pragma unroll
        for (int r = 0; r < 8; ++r) {
            long  row = rowbase + r + half * 8;       // C/D layout: M = r + 8*half
            float Ih  = accI[j][r];
            float e0  = hid_E0[row * 64 + ng];
            float Eh  = fmaf(a4c, e0, sc * Ih);
            o_Eh[row * 64 + ng] = Eh;
            o_Oh[row * 64 + ng] = (Eh >= thc) ? Eh : 0.0f;
        }
    }

    // ---- Output epilogue (cols 8..15 of the padded tile are dead) ----
    if (n < 8) {
        const float dco = dec_out[n];
        const float tho = th_out[n];
        #pragma unroll
        for (int r = 0; r < 8; ++r) {
            long  row = rowbase + r + half * 8;
            float e0  = out_E0[row * 8 + n];
            float Eo  = fmaf(dco, e0, accO[r]);
            o_Eo[row * 8 + n]  = Eo;
            o_Oo[row * 8 + n]  = (Eo >= tho) ? Eo : 0.0f;
            o_act[row * 8 + n] = tanhf(Eo);
        }
    }
}

// ---------------------------------------------------------------------------
extern "C" void kernel_launch(void* const* d_in, const int* in_sizes, int n_in,
                              void* d_out, int out_size, void* d_ws, size_t ws_size,
                              hipStream_t stream) {
    const float* x         = (const float*)d_in[0];
    const float* hid_E0    = (const float*)d_in[1];
    const float* hid_O0    = (const float*)d_in[2];
    const float* out_E0    = (const float*)d_in[3];
    /* d_in[4] = out_O0 : unused by reference */
    const float* w_in2hid  = (const float*)d_in[5];
    const float* w_hid_IN  = (const float*)d_in[6];
    const float* w_hid_EX  = (const float*)d_in[7];
    const float* w_hid2out = (const float*)d_in[8];
    const float* gj_w      = (const float*)d_in[9];
    const float* th_in     = (const float*)d_in[10];
    const float* dec_in    = (const float*)d_in[11];
    const float* th_hid    = (const float*)d_in[12];
    const float* dec_hid   = (const float*)d_in[13];
    const float* th_out    = (const float*)d_in[14];
    const float* dec_out   = (const float*)d_in[15];
    const int*   gj_src    = (const int*)d_in[16];
    const int*   gj_dst    = (const int*)d_in[17];

    float* ws  = (float*)d_ws;
    float* out = (float*)d_out;
    const int B = in_sizes[0] / 4;

    twc_prep<<<1, 256, 0, stream>>>(w_in2hid, w_hid_IN, w_hid_EX, w_hid2out,
                                    gj_w, dec_hid, th_hid, gj_src, gj_dst, ws);

    const int blocks = (B + 127) / 128;   // 8 waves/block * 16 rows/wave
    twc_main<<<blocks, 256, 0, stream>>>(x, hid_E0, hid_O0, out_E0,
                                         th_in, dec_in, th_out, dec_out,
                                         ws, out, B);
}